// LocalEntropyGating_84078279786766
// MI455X (gfx1250) — compile-verified
//
#include <hip/hip_runtime.h>
#include <math.h>

// LocalEntropyGating for MI455X (gfx1250, wave32).
// x: [32,3,512,512] f32 -> mask: [32,1,512,512] f32.
// Phase 1: per-wave 16x16 tile; 21x21 box sums of gray and gray^2 computed as
//          banded-matrix WMMA matmuls (f32 16x16x4), std -> d_out, min/max -> ws.
// Phase 2: elementwise normalize + sigmoid in place on d_out.

typedef float v2f __attribute__((ext_vector_type(2)));
typedef float v8f __attribute__((ext_vector_type(8)));

#define NB   32
#define HW   512
#define IMG  (HW * HW)

__device__ __forceinline__ v8f wmma_f32_16x16x4(v2f a, v2f b, v8f c) {
#if defined(__gfx1250__) && __has_builtin(__builtin_amdgcn_wmma_f32_16x16x4_f32)
  // (neg_a, A, neg_b, B, c_mod, C, reuse_a, reuse_b)
  return __builtin_amdgcn_wmma_f32_16x16x4_f32(false, a, false, b, (short)0, c,
                                               false, false);
#else
  // Host pass / builtin-missing fallback (keeps compile clean; histogram will
  // show wmma==0 if this path was taken on device).
  c[0] += a[0] * b[0];
  return c;
#endif
}

__global__ void init_minmax(unsigned* __restrict__ wsmm) {
  if (threadIdx.x == 0) {
    wsmm[0] = 0x7F800000u;  // +inf bits: accumulator for uint atomicMin
    wsmm[1] = 0u;           // accumulator for uint atomicMax (std > 0 always)
  }
}

// Block = 128 threads = 4 waves; each wave owns one 16x16 output tile.
// Per-wave LDS: Gb = 48x49 grayscale patch (zero-padded halo), Tb = 48x17
// horizontal-sum buffer reused as the final 16x17 transpose staging area.
__global__ __launch_bounds__(128) void std_kernel(const float* __restrict__ x,
                                                  float* __restrict__ out,
                                                  unsigned* __restrict__ wsmm) {
  __shared__ float lds[4 * (48 * 49 + 48 * 17)];  // 50688 B
  const int wid  = threadIdx.x >> 5;
  const int lane = threadIdx.x & 31;
  float* Gb = lds + wid * (48 * 49 + 48 * 17);
  float* Tb = Gb + 48 * 49;

  const int tile = blockIdx.x * 4 + wid;   // 32768 tiles total
  const int b  = tile >> 10;               // image
  const int It = (tile >> 5) & 31;         // row tile
  const int Jt = tile & 31;                // col tile

  // ---- Stage 48x48 grayscale patch (zero pad outside image), coalesced ----
  const float* xb = x + (size_t)b * 3 * IMG;
  for (int i = 0; i < 72; ++i) {           // 72*32 = 2304 = 48*48
    int idx = lane + i * 32;
    int r  = idx / 48;
    int cc = idx - r * 48;
    int gy = It * 16 - 16 + r;
    int gx = Jt * 16 - 16 + cc;
    float v = 0.0f;
    if ((unsigned)gy < (unsigned)HW && (unsigned)gx < (unsigned)HW) {
      int o = gy * HW + gx;
      v = fmaf(0.299f, xb[o],
          fmaf(0.587f, xb[IMG + o], 0.114f * xb[2 * IMG + o]));
    }
    Gb[r * 49 + cc] = v;
  }
  __syncthreads();

  const int m    = lane & 15;   // N-column (and A row M)
  const int half = lane >> 4;   // lane half selects K sub-pair in A/B layouts

  // ---- Band-matrix B blocks, generated in-register (shared by both passes) --
  // B 4x16 layout: VGPR0 = K0 | K2, VGPR1 = K1 | K3 across lane halves.
  // Bm[c][g] covers k16 = 4g..4g+3 of the 16x16 band block with offset (c-1)*16:
  //   W[k16][n] = 1  iff  |16*(c-1) + k16 - n| <= 10
  v2f Bm[3][4];
#pragma unroll
  for (int c = 0; c < 3; ++c)
#pragma unroll
    for (int g = 0; g < 4; ++g) {
      int d0 = (c - 1) * 16 + g * 4 + 2 * half - m;
      Bm[c][g][0] = (d0 >= -10 && d0 <= 10) ? 1.0f : 0.0f;
      Bm[c][g][1] = (d0 + 1 >= -10 && d0 + 1 <= 10) ? 1.0f : 0.0f;
    }

  v8f S = {0, 0, 0, 0, 0, 0, 0, 0};
  v8f S2 = {0, 0, 0, 0, 0, 0, 0, 0};

#pragma unroll
  for (int sq = 0; sq < 2; ++sq) {
    // ---- Horizontal pass: T_R = sum_C G_{R,C} * Wh_{C-1}, R = 0..2 ----------
#pragma unroll
    for (int R = 0; R < 3; ++R) {
      v8f acc = {0, 0, 0, 0, 0, 0, 0, 0};
#pragma unroll
      for (int C = 0; C < 3; ++C)
#pragma unroll
        for (int g = 0; g < 4; ++g) {
          // A 16x4 layout: M = lane&15 in both halves; K = 4g + j + 2*half
          const float* p = &Gb[(R * 16 + m) * 49 + C * 16 + g * 4 + 2 * half];
          v2f a = {p[0], p[1]};
          if (sq) { a[0] *= a[0]; a[1] *= a[1]; }
          acc = wmma_f32_16x16x4(a, Bm[C][g], acc);
        }
      // D layout: VGPR v -> row (v + 8*half), col m. Store T rows to LDS.
#pragma unroll
      for (int v = 0; v < 8; ++v)
        Tb[(R * 16 + v + 8 * half) * 17 + m] = acc[v];
    }
    __syncthreads();

    // ---- Vertical pass on the transpose: S^T = sum_R T^T_R * Wv_{R-1} ------
    v8f s = {0, 0, 0, 0, 0, 0, 0, 0};
#pragma unroll
    for (int R = 0; R < 3; ++R)
#pragma unroll
      for (int g = 0; g < 4; ++g) {
        int r = R * 16 + g * 4 + 2 * half;       // K = input row index
        v2f a = {Tb[r * 17 + m], Tb[(r + 1) * 17 + m]};  // A[m=x'][k=row]
        s = wmma_f32_16x16x4(a, Bm[R][g], s);
      }
    if (sq == 0) S = s; else S2 = s;
    __syncthreads();  // WAR: Tb reused by next iteration / transpose staging
  }

  // ---- std = sqrt(clip(mu2 - mu^2, 1e-6)); local min/max -------------------
  const float inv = 1.0f / 441.0f;
  float stdv[8];
  float mn = 3.4e38f, mx = -3.4e38f;
#pragma unroll
  for (int v = 0; v < 8; ++v) {
    float mu  = S[v] * inv;
    float var = fmaxf(S2[v] * inv - mu * mu, 1e-6f);
    float st  = sqrtf(var);
    stdv[v] = st;
    mn = fminf(mn, st);
    mx = fmaxf(mx, st);
  }

  // S^T D-layout element = (x' = v + 8*half, y' = m): stage as Out[y'][x'].
#pragma unroll
  for (int v = 0; v < 8; ++v)
    Tb[m * 17 + v + 8 * half] = stdv[v];
  __syncthreads();

  // ---- Coalesced 16x16 tile store: 2 lanes per row, float4 x2 per lane -----
  {
    int y  = lane >> 1;
    int xh = (lane & 1) * 8;
    const float* p = &Tb[y * 17 + xh];
    float4 lo = make_float4(p[0], p[1], p[2], p[3]);
    float4 hi = make_float4(p[4], p[5], p[6], p[7]);
    float* op = out + (size_t)b * IMG + (It * 16 + y) * HW + Jt * 16 + xh;
    reinterpret_cast<float4*>(op)[0] = lo;
    reinterpret_cast<float4*>(op)[1] = hi;
  }

  // ---- Wave reduce min/max, one atomic pair per wave (uint-ordered floats) -
#pragma unroll
  for (int off = 16; off; off >>= 1) {
    mn = fminf(mn, __shfl_xor(mn, off));
    mx = fmaxf(mx, __shfl_xor(mx, off));
  }
  if (lane == 0) {
    atomicMin(&wsmm[0], __float_as_uint(mn));
    atomicMax(&wsmm[1], __float_as_uint(mx));
  }
}

// Phase 2: mask = sigmoid(10 * ((std - min)/(max - min + 1e-6) - 0.2)), in place.
__global__ __launch_bounds__(256) void norm_kernel(float* __restrict__ out,
                                                   const unsigned* __restrict__ wsmm) {
  const float smin = __uint_as_float(wsmm[0]);
  const float smax = __uint_as_float(wsmm[1]);
  const float sc = 1.0f / (smax - smin + 1e-6f);
  int idx = blockIdx.x * 256 + threadIdx.x;  // 2,097,152 float4s exactly
  float4* o4 = reinterpret_cast<float4*>(out);
  float4 v = o4[idx];
  float* pv = reinterpret_cast<float*>(&v);
#pragma unroll
  for (int i = 0; i < 4; ++i) {
    float t = (pv[i] - smin) * sc;
    pv[i] = 1.0f / (1.0f + __expf(-10.0f * (t - 0.2f)));
  }
  o4[idx] = v;
}

extern "C" void kernel_launch(void* const* d_in, const int* in_sizes, int n_in,
                              void* d_out, int out_size, void* d_ws, size_t ws_size,
                              hipStream_t stream) {
  (void)in_sizes; (void)n_in; (void)out_size; (void)ws_size;
  const float* x   = (const float*)d_in[0];
  float* out       = (float*)d_out;
  unsigned* wsmm   = (unsigned*)d_ws;

  init_minmax<<<1, 32, 0, stream>>>(wsmm);
  // 32 images * 32 * 32 tiles = 32768 wave-tiles; 4 waves per 128-thread block.
  std_kernel<<<(NB * 32 * 32) / 4, 128, 0, stream>>>(x, out, wsmm);
  // 32*512*512 / 4 elements per thread / 256 threads = 8192 blocks.
  norm_kernel<<<(NB * IMG) / (4 * 256), 256, 0, stream>>>(out, wsmm);
}